// VectorQuantizerEMA_77936476553962
// MI455X (gfx1250) — compile-verified
//
#include <hip/hip_runtime.h>
#include <hip/hip_bf16.h>

typedef __attribute__((ext_vector_type(2))) float v2f;
typedef __attribute__((ext_vector_type(8))) float v8f;
typedef unsigned int u32x4 __attribute__((ext_vector_type(4)));
typedef int          i32x4 __attribute__((ext_vector_type(4)));
typedef int          i32x8 __attribute__((ext_vector_type(8)));

#define VQ_K      1024
#define VQ_D      64
#define VQ_HW     4096          // 64*64
#define VQ_N      131072        // 32*64*64
#define VQ_TILES  8192          // VQ_N / 16
#define OUT_QUANT 8388608       // 32*64*64*64

// Toolchain arity shim for the TDM builtin (6-arg therock vs 5-arg ROCm).
#if __has_include(<hip/amd_detail/amd_gfx1250_TDM.h>)
#define VQ_TDM_LOAD(d0, d1, z4, z8) \
    __builtin_amdgcn_tensor_load_to_lds((d0), (d1), (z4), (z4), (z8), 0)
#else
#define VQ_TDM_LOAD(d0, d1, z4, z8) \
    __builtin_amdgcn_tensor_load_to_lds((d0), (d1), (z4), (z4), 0)
#endif

// ---------------- workspace zero ----------------
__global__ void vq_zero_kernel(float* p, int n) {
    int i = blockIdx.x * blockDim.x + threadIdx.x;
    if (i < n) p[i] = 0.0f;
}

// ---------------- codebook squared norms ----------------
__global__ void vq_enorm_kernel(const float* __restrict__ embed,
                                float* __restrict__ enorm) {
    int k = blockIdx.x * blockDim.x + threadIdx.x;
    if (k < VQ_K) {
        const float* e = embed + (size_t)k * VQ_D;
        float s = 0.0f;
        #pragma unroll
        for (int c = 0; c < VQ_D; ++c) { float v = e[c]; s += v * v; }
        enorm[k] = s;
    }
}

// ---------------- pre-swizzle codebook into WMMA B-fragment order ----------------
// swz[ct*512 + kk*32 + lane] = float2 that lane `lane` feeds to WMMA at k-step kk
// for code tile ct:  embed[ct*16 + lane%16][4*kk + 2*(lane/16) + {0,1}]
__global__ void vq_bswizzle_kernel(const float* __restrict__ embed,
                                   v2f* __restrict__ swz) {
    int t = blockIdx.x * blockDim.x + threadIdx.x;     // 0 .. 64*512-1
    if (t < 64 * 512) {
        const int ct  = t >> 9;
        const int rem = t & 511;
        const int kk  = rem >> 5;
        const int l   = rem & 31;
        const int r   = l & 15;
        const int kh  = l >> 4;
        const int c0  = 4 * kk + 2 * kh;
        const float* e = embed + (size_t)(ct * 16 + r) * VQ_D + c0;
        v2f v; v.x = e[0]; v.y = e[1];
        swz[t] = v;
    }
}

// ---------------- main kernel: TDM-fed WMMA distances + argmin + quantize + stats ----------------
__global__ __launch_bounds__(256) void vq_argmin_kernel(
    const float* __restrict__ x,       // [32,64,64,64] NCHW
    const float* __restrict__ embed,   // [1024,64]
    const float* __restrict__ enorm,   // [1024]
    const v2f*   __restrict__ swz,     // [64 tiles][16 kk][32 lanes] B fragments
    float* __restrict__ quant_out,     // [32,64,64,64] NCHW
    float* __restrict__ idx_out,       // [131072] (indices stored as float)
    float* __restrict__ counts,        // [1024]
    float* __restrict__ dw,            // [1024*64]
    float* __restrict__ sse_acc)       // [1]
{
    // Double-buffered B-fragment staging (4 KB per code tile), filled by TDM.
    __shared__ v2f bbuf[2][512];
    __shared__ int s_idx[8][16];

    const int lane = threadIdx.x & 31;
    const int wave = threadIdx.x >> 5;
    const int tile = blockIdx.x * 8 + wave;        // 1024 blocks * 8 waves
    const int n0   = tile * 16;
    const int b    = n0 >> 12;                     // / 4096
    const int rem  = n0 & 4095;
    const int h    = rem >> 6;
    const int w0   = rem & 63;                     // multiple of 16
    const int kh   = lane >> 4;                    // half-wave id (0/1)
    const int lr   = lane & 15;                    // row (A) / code column (B)

    // element x[b][c][h][w0+lr]: base with c=0
    const size_t xbase = (size_t)b * (VQ_D * VQ_HW) + (size_t)h * 64 + w0 + lr;

    // Load entire 16x64 x-tile into A fragments (WMMA 16x16x4 f32 layout).
    v2f a[16];
    #pragma unroll
    for (int kk = 0; kk < 16; ++kk) {
        const int c0 = 4 * kk + 2 * kh;
        a[kk].x = x[xbase + (size_t)c0 * VQ_HW];
        a[kk].y = x[xbase + (size_t)(c0 + 1) * VQ_HW];
    }

    // TDM descriptor pieces (uniform). Low 32 bits of a flat LDS pointer are the
    // workgroup-relative LDS byte offset (ISA 10.2 aperture mapping).
    const unsigned lds_off[2] = {
        (unsigned)(unsigned long long)(&bbuf[0][0]),
        (unsigned)(unsigned long long)(&bbuf[1][0])
    };
    const unsigned long long gbase = (unsigned long long)swz;
    const i32x4 z4 = {};
    const i32x8 z8 = {};

    auto tdm_issue = [&](int ct, int sel) {
        const unsigned long long ga = gbase + (unsigned long long)ct * 4096ull;
        u32x4 d0 = {};
        d0[0] = 1u;                                   // count=1, user descriptor
        d0[1] = lds_off[sel];                         // lds_addr (bytes)
        d0[2] = (unsigned)ga;                         // global_addr[31:0]
        d0[3] = (unsigned)(ga >> 32) | (2u << 30);    // global_addr[56:32] | type=2
        i32x8 d1 = {};
        d1[0] = (int)(3u << 16);                      // data_size = 8 bytes
        d1[1] = (int)(512u << 16);                    // tensor_dim0 = 512 units
        d1[2] = (int)(1u << 16);                      // tensor_dim1 = 1
        d1[3] = (int)(512u << 16);                    // tile_dim0 = 512 units (4 KB)
        d1[4] = 1;                                    // tile_dim1 = 1
        d1[5] = 512;                                  // tensor_dim0_stride = 512
        d1[6] = (int)(512u << 16);                    // tensor_dim1_stride = 512
        d1[7] = 0;
        VQ_TDM_LOAD(d0, d1, z4, z8);
    };

    // Prologue: stage code tile 0.
    if (wave == 0) {
        tdm_issue(0, 0);
        __builtin_amdgcn_s_wait_tensorcnt(0);
    }
    __syncthreads();

    float minval[8];
    int   minidx[8];
    #pragma unroll
    for (int v = 0; v < 8; ++v) { minval[v] = 3.4e38f; minidx[v] = 0; }

    // Sweep the 64 code tiles; TDM prefetches ct+1 while all 8 waves WMMA ct.
    for (int ct = 0; ct < 64; ++ct) {
        if (wave == 0 && ct + 1 < 64) tdm_issue(ct + 1, (ct + 1) & 1);

        const v2f* bp = &bbuf[ct & 1][lane];          // conflict-free ds_load_b64
        v8f acc = {};
        #pragma unroll
        for (int kk = 0; kk < 16; ++kk) {
            const v2f bf = bp[kk * 32];
            acc = __builtin_amdgcn_wmma_f32_16x16x4_f32(
                false, a[kk], false, bf, (short)0, acc, false, false);
        }

        const int cb = ct * 16;
        const float en = enorm[cb + lr];
        #pragma unroll
        for (int v = 0; v < 8; ++v) {
            // row = v + 8*kh, code = cb + lr ; ||x||^2 dropped (argmin-invariant)
            const float dist = en - 2.0f * acc[v];
            if (dist < minval[v]) { minval[v] = dist; minidx[v] = cb + lr; }
        }

        if (wave == 0 && ct + 1 < 64) __builtin_amdgcn_s_wait_tensorcnt(0);
        __syncthreads();
    }

    // Argmin reduction across the 16 lanes of each half-wave (offsets < 16
    // never cross the half boundary). Tie-break: smaller index.
    #pragma unroll
    for (int v = 0; v < 8; ++v) {
        float val = minval[v];
        int   idx = minidx[v];
        #pragma unroll
        for (int off = 8; off > 0; off >>= 1) {
            const float oval = __shfl_xor(val, off);
            const int   oidx = __shfl_xor(idx, off);
            if (oval < val || (oval == val && oidx < idx)) { val = oval; idx = oidx; }
        }
        minval[v] = val; minidx[v] = idx;
    }

    // lane lr==0 of each half holds rows v + 8*kh
    if (lr == 0) {
        #pragma unroll
        for (int v = 0; v < 8; ++v) s_idx[wave][v + 8 * kh] = minidx[v];
    }
    __syncthreads();

    const int k_r = s_idx[wave][lr];               // code for row lr of this tile
    if (kh == 0) {
        atomicAdd(&counts[k_r], 1.0f);
        idx_out[n0 + lr] = (float)k_r;
    }

    // Quantize, straight-through output, SSE, and dw scatter.
    const float* qrow = embed + (size_t)k_r * VQ_D;
    float sse = 0.0f;
    #pragma unroll
    for (int kk = 0; kk < 16; ++kk) {
        const int c0 = 4 * kk + 2 * kh;
        const float x0 = a[kk].x, x1 = a[kk].y;
        const float q0 = qrow[c0], q1 = qrow[c0 + 1];
        const float d0 = q0 - x0, d1 = q1 - x1;
        quant_out[xbase + (size_t)c0 * VQ_HW]       = x0 + d0;   // x + (q - x)
        quant_out[xbase + (size_t)(c0 + 1) * VQ_HW] = x1 + d1;
        sse += d0 * d0 + d1 * d1;
        atomicAdd(&dw[k_r * VQ_D + c0],     x0);
        atomicAdd(&dw[k_r * VQ_D + c0 + 1], x1);
    }

    // wave-level SSE reduction, one atomic per wave
    #pragma unroll
    for (int off = 16; off > 0; off >>= 1) sse += __shfl_xor(sse, off);
    if (lane == 0) atomicAdd(sse_acc, sse);
}

// ---------------- finalize: EMA update, embed refresh, loss, perplexity ----------------
__global__ __launch_bounds__(1024) void vq_finalize_kernel(
    const float* __restrict__ ema_cs,  // [1024]
    const float* __restrict__ ema_w,   // [1024,64]
    const float* __restrict__ counts,  // [1024]
    const float* __restrict__ dw,      // [1024,64]
    const float* __restrict__ sse,     // [1]
    float* __restrict__ out_loss,
    float* __restrict__ out_perp,
    float* __restrict__ out_ncs,       // [1024]
    float* __restrict__ out_nw,        // [1024,64]
    float* __restrict__ out_nembed)    // [1024,64]
{
    __shared__ float red_n[1024];
    __shared__ float red_e[1024];
    const int k = threadIdx.x;

    const float cnt = counts[k];
    const float ncs = 0.99f * ema_cs[k] + 0.01f * cnt;
    out_ncs[k] = ncs;

    const float p = cnt / (float)VQ_N;
    red_n[k] = ncs;
    red_e[k] = p * logf(p + 1e-10f);
    __syncthreads();
    for (int s = 512; s > 0; s >>= 1) {
        if (k < s) { red_n[k] += red_n[k + s]; red_e[k] += red_e[k + s]; }
        __syncthreads();
    }
    const float n = red_n[0];

    float cs = (ncs + 1e-5f) / (n + (float)VQ_K * 1e-5f) * n;
    cs = fmaxf(cs, 1e-5f);

    const float* wrow = ema_w + (size_t)k * VQ_D;
    const float* drow = dw    + (size_t)k * VQ_D;
    #pragma unroll 4
    for (int c = 0; c < VQ_D; ++c) {
        const float nw = 0.99f * wrow[c] + 0.01f * drow[c];
        out_nw[(size_t)k * VQ_D + c]     = nw;
        out_nembed[(size_t)k * VQ_D + c] = nw / cs;
    }

    if (k == 0) {
        *out_loss = 0.25f * (*sse) / (float)OUT_QUANT;   // COMMITMENT_COST * mean
        *out_perp = expf(-red_e[0]);
    }
}

extern "C" void kernel_launch(void* const* d_in, const int* in_sizes, int n_in,
                              void* d_out, int out_size, void* d_ws, size_t ws_size,
                              hipStream_t stream) {
    const float* x      = (const float*)d_in[0];   // inputs
    const float* embed  = (const float*)d_in[1];   // codebook
    const float* ema_cs = (const float*)d_in[2];
    const float* ema_w  = (const float*)d_in[3];

    float* out      = (float*)d_out;
    float* quant    = out;                                  // 8388608
    float* loss     = out + OUT_QUANT;                      // 1
    float* perp     = out + OUT_QUANT + 1;                  // 1
    float* idx_out  = out + OUT_QUANT + 2;                  // 131072
    float* ncs      = idx_out + VQ_N;                       // 1024
    float* nw       = ncs + VQ_K;                           // 65536
    float* nembed   = nw + VQ_K * VQ_D;                     // 65536

    float* ws      = (float*)d_ws;
    v2f*   swz     = (v2f*)ws;                              // 32768 v2f = 65536 f
    float* enorm   = ws + 65536;                            // 1024
    float* counts  = ws + 65536 + VQ_K;                     // 1024
    float* dwbuf   = ws + 65536 + 2 * VQ_K;                 // 65536
    float* sse     = dwbuf + VQ_K * VQ_D;                   // 1
    const int zero_floats = 2 * VQ_K + VQ_K * VQ_D + 1;     // counts+dw+sse

    vq_zero_kernel<<<(zero_floats + 255) / 256, 256, 0, stream>>>(counts, zero_floats);
    vq_enorm_kernel<<<VQ_K / 256, 256, 0, stream>>>(embed, enorm);
    vq_bswizzle_kernel<<<(64 * 512) / 256, 256, 0, stream>>>(embed, swz);
    vq_argmin_kernel<<<VQ_TILES / 8, 256, 0, stream>>>(
        x, embed, enorm, swz, quant, idx_out, counts, dwbuf, sse);
    vq_finalize_kernel<<<1, 1024, 0, stream>>>(
        ema_cs, ema_w, counts, dwbuf, sse, loss, perp, ncs, nw, nembed);
}